// AddNet_403726926589
// MI455X (gfx1250) — compile-verified
//
#include <hip/hip_runtime.h>
#include <stdint.h>

// AdderNet depthwise 3x3 L1-distance conv, stride 1, pad 1, no bias.
//   out[b,c,h,w] = - sum_{i,j} | x_pad[b,c,h+i,w+j] - W[c,0,i,j] |
// x: (32,4,512,512) f32, W: (4,1,3,3) f32.

#define IMG_H 512
#define IMG_W 512
#define NCH   4
#define NB    32

#define TILE_W     128          // output tile width per block
#define TILE_H     32           // output tile height per block
#define LDS_COLS   136          // TILE_W + 8 (aligned halo: global col bx*128-4 .. +131)
#define LDS_ROWS   34           // TILE_H + 2
#define SEG_PER_ROW 34          // 136 floats / 4 per b128 segment
#define NSEG       (LDS_ROWS * SEG_PER_ROW)   // 1156 segments
#define NTHREADS   256

typedef int   v4i  __attribute__((ext_vector_type(4)));
typedef float f32x4 __attribute__((ext_vector_type(4)));

typedef __attribute__((address_space(1))) v4i g_v4i;   // global (AS1)
typedef __attribute__((address_space(3))) v4i l_v4i;   // LDS (AS3)

// ---------------- CDNA5 async global->LDS helpers ----------------

__device__ __forceinline__ void async_copy_b128_g2l(const float* gsrc, float* ldst) {
#if __has_builtin(__builtin_amdgcn_global_load_async_to_lds_b128)
  __builtin_amdgcn_global_load_async_to_lds_b128(
      (g_v4i*)gsrc, (l_v4i*)ldst, /*offset=*/0, /*cpol=*/0);
#else
  // VDST = VGPR with LDS byte address, VADDR = 64-bit global address, no SADDR.
  asm volatile("global_load_async_to_lds_b128 %0, %1, off"
               :
               : "v"((l_v4i*)ldst), "v"((g_v4i*)gsrc)
               : "memory");
#endif
}

__device__ __forceinline__ void wait_asynccnt_zero() {
#if __has_builtin(__builtin_amdgcn_s_wait_asynccnt)
  __builtin_amdgcn_s_wait_asynccnt(0);
#else
  asm volatile("s_wait_asynccnt 0" ::: "memory");
#endif
}

// ---------------- kernel ----------------

__global__ __launch_bounds__(NTHREADS)
void addernet_dw3x3(const float* __restrict__ x,
                    const float* __restrict__ wgt,
                    float* __restrict__ out) {
  __shared__ float tile[LDS_ROWS * LDS_COLS];   // 34*136*4 = 18,496 B of 320 KB

  const int bx = blockIdx.x;                    // tile col  [0,4)
  const int by = blockIdx.y;                    // tile row  [0,16)
  const int p  = blockIdx.z;                    // plane b*NCH + c, [0,128)
  const int c  = p & (NCH - 1);

  const float* __restrict__ plane  = x   + (size_t)p * IMG_H * IMG_W;
  float* __restrict__       oplane = out + (size_t)p * IMG_H * IMG_W;

  const int gxbase = bx * TILE_W - 4;           // global col of LDS col 0 (16B aligned)
  const int gy0    = by * TILE_H - 1;           // global row of LDS row 0
  const int tid    = threadIdx.x;

  // ---- stage halo tile as aligned b128 segments: async load or zero-fill ----
  for (int seg = tid; seg < NSEG; seg += NTHREADS) {
    const int row  = seg / SEG_PER_ROW;         // LDS row      [0,34)
    const int s    = seg - row * SEG_PER_ROW;   // segment in row [0,34)
    const int gy   = gy0 + row;
    const int gxs  = gxbase + s * 4;            // first global col of segment
    float* ldst = &tile[row * LDS_COLS + s * 4];
    // segments are 4-float aligned: either fully in-image or fully outside
    if ((unsigned)gy < (unsigned)IMG_H && (unsigned)gxs < (unsigned)(IMG_W - 3)) {
      async_copy_b128_g2l(plane + (size_t)gy * IMG_W + gxs, ldst);
    } else {
      *(f32x4*)ldst = (f32x4)0.0f;              // zero padding (ds_store_b128)
    }
  }
  wait_asynccnt_zero();                         // drain this wave's ASYNCcnt
  __syncthreads();                              // cross-wave LDS visibility

  // ---- per-channel taps: uniform index -> scalar loads ----
  float wk[9];
#pragma unroll
  for (int t = 0; t < 9; ++t) wk[t] = wgt[c * 9 + t];

  // ---- each thread: 4x4 outputs from a 6x6 LDS window (2.25 LDS reads/out) ----
  const int tx = tid & 31;                      // [0,32)
  const int ty = tid >> 5;                      // [0,8)
  const int oy = ty * 4;                        // output row within tile
  const int ox = tx * 4;                        // output col within tile
  // LDS col of leftmost tap for output col ox: (ox + 4) - 1 = ox + 3

  float win[6][6];
#pragma unroll
  for (int r = 0; r < 6; ++r)
#pragma unroll
    for (int q = 0; q < 6; ++q)
      win[r][q] = tile[(oy + r) * LDS_COLS + (ox + 3 + q)];

  float acc[4][4];
#pragma unroll
  for (int r = 0; r < 4; ++r)
#pragma unroll
    for (int q = 0; q < 4; ++q)
      acc[r][q] = 0.0f;

#pragma unroll
  for (int i = 0; i < 3; ++i) {
#pragma unroll
    for (int j = 0; j < 3; ++j) {
      const float wv = wk[i * 3 + j];
#pragma unroll
      for (int r = 0; r < 4; ++r)
#pragma unroll
        for (int q = 0; q < 4; ++q)
          acc[r][q] -= fabsf(win[r + i][q + j] - wv);  // v_sub + v_sub |src|
    }
  }

  // ---- coalesced non-temporal float4 stores (output is write-once) ----
  const int gyo = by * TILE_H + oy;
  const int gxo = bx * TILE_W + ox;
#pragma unroll
  for (int r = 0; r < 4; ++r) {
    f32x4 v = {acc[r][0], acc[r][1], acc[r][2], acc[r][3]};
    __builtin_nontemporal_store(v, (f32x4*)(oplane + (size_t)(gyo + r) * IMG_W + gxo));
  }
}

// ---------------- launcher ----------------

extern "C" void kernel_launch(void* const* d_in, const int* in_sizes, int n_in,
                              void* d_out, int out_size, void* d_ws, size_t ws_size,
                              hipStream_t stream) {
  const float* x = (const float*)d_in[0];   // (32,4,512,512) f32
  const float* w = (const float*)d_in[1];   // (4,1,3,3) f32
  float* out = (float*)d_out;               // (32,4,512,512) f32

  dim3 grid(IMG_W / TILE_W, IMG_H / TILE_H, NB * NCH);   // (4, 16, 128)
  addernet_dw3x3<<<grid, NTHREADS, 0, stream>>>(x, w, out);
}